// GCN_skip_2_layer_linear_in_out_7567732376252
// MI455X (gfx1250) — compile-verified
//
#include <hip/hip_runtime.h>
#include <hip/hip_bf16.h>

#define N_NODES 50000
#define N_EDGES 600000
#define HID 128
#define OUT_C 64
#define N_GCN 4
#define NF (N_NODES * HID)   // 6,400,000 floats per node-feature buffer
#define LDA 132              // padded LDS row stride: banks (4l+kb+2h) -> conflict-free

typedef float v2f __attribute__((ext_vector_type(2)));
typedef float v8f __attribute__((ext_vector_type(8)));

// ---------------------------------------------------------------------------
// Generic fill
// ---------------------------------------------------------------------------
__global__ void k_fill(float* __restrict__ p, int n, float v) {
    int i = blockIdx.x * blockDim.x + threadIdx.x;
    if (i < n) p[i] = v;
}

// deg[dst[e]] += 1  (deg buffer pre-zeroed)
__global__ void k_deg(float* __restrict__ deg, const int* __restrict__ dst, int n_edges) {
    int e = blockIdx.x * blockDim.x + threadIdx.x;
    if (e < n_edges) atomicAdd(&deg[dst[e]], 1.0f);
}

// in-place: dis[n] = rsqrt(deg[n] + 1)   (self-loop)
__global__ void k_dis(float* __restrict__ d, int n) {
    int i = blockIdx.x * blockDim.x + threadIdx.x;
    if (i < n) d[i] = __frsqrt_rn(d[i] + 1.0f);
}

// ---------------------------------------------------------------------------
// fp32 WMMA GEMM:  C[M x NCOLS] = A[M x 128] * B[128 x NCOLS] (+ bias)
// One wave computes a 16x16 output tile via V_WMMA_F32_16X16X4_F32.
// Block = (NCOLS/16) waves covering 16 rows x NCOLS cols; grid.x = M/16.
// M = 50000 = 3125*16 exactly -> no tail, EXEC all ones (WMMA requirement).
//
// A tile (16 rows x 128 K) is staged once per block into LDS with coalesced
// float2 loads (shared by all waves), then fragments come from ds_load_b64
// at conflict-free padded addresses. B fragments stream from global (64 KB,
// L1/L2 resident across all 3125 blocks).
//
// A frag (16x4, 2 VGPRs): lane l(0..15) half h: a[v] = A[row0+l][kb + 2h + v]
// B frag (4x16, 2 VGPRs): lane l half h:        b[v] = B[kb + 2h + v][col0+l]
// D (16x16, 8 VGPRs): vgpr r, half h: D[row0 + r + 8h][col0 + l]
// ---------------------------------------------------------------------------
template <int NCOLS>
__global__ __launch_bounds__(NCOLS * 2)
void k_gemm(float* __restrict__ C, const float* __restrict__ A,
            const float* __restrict__ B, const float* __restrict__ bias) {
    __shared__ float As[16 * LDA];

    const int tid  = threadIdx.x;
    const int row0 = blockIdx.x * 16;

    // Cooperative A-tile stage: 16 rows x 64 float2 = 1024 chunks.
#pragma unroll
    for (int idx = tid; idx < 16 * 64; idx += NCOLS * 2) {
        const int r  = idx >> 6;
        const int c2 = idx & 63;
        const float2 v = *(const float2*)(A + (size_t)(row0 + r) * HID + 2 * c2);
        *(float2*)(As + r * LDA + 2 * c2) = v;
    }
    __syncthreads();

    const int wave = tid >> 5;
    const int lane = tid & 31;
    const int half = lane >> 4;
    const int l    = lane & 15;
    const int col0 = wave * 16;

    const float* __restrict__ Al = As + l * LDA + 2 * half;

    v8f c = {};
#pragma unroll
    for (int kb = 0; kb < HID; kb += 4) {
        v2f a, b;
        const float2 av = *(const float2*)(Al + kb);   // ds_load_b64, bank-conflict-free
        a.x = av.x;
        a.y = av.y;
        const float* bp = B + (size_t)(kb + 2 * half) * NCOLS + col0 + l;
        b.x = bp[0];
        b.y = bp[NCOLS];
        c = __builtin_amdgcn_wmma_f32_16x16x4_f32(
            /*neg_a=*/false, a, /*neg_b=*/false, b,
            /*c_mod=*/(short)0, c, /*reuse_a=*/false, /*reuse_b=*/false);
    }

    const float bv = bias ? bias[col0 + l] : 0.0f;
#pragma unroll
    for (int r = 0; r < 8; ++r) {
        C[(size_t)(row0 + r + 8 * half) * NCOLS + col0 + l] = c[r] + bv;
    }
}

// ---------------------------------------------------------------------------
// agg = t * self_norm + b_g[i]      (float4-vectorized; 32 float4 per row)
// ---------------------------------------------------------------------------
__global__ void k_agg_init(float4* __restrict__ agg4, const float4* __restrict__ t4,
                           const float* __restrict__ dis, const float* __restrict__ bg,
                           int total4) {
    int i = blockIdx.x * blockDim.x + threadIdx.x;
    if (i >= total4) return;
    int n = i >> 5;      // row  (HID/4 = 32 float4 per row)
    int c = i & 31;      // float4 column
    float sn = dis[n];
    sn = sn * sn;
    float4 tv = t4[i];
    float4 bv = ((const float4*)bg)[c];
    float4 o;
    o.x = tv.x * sn + bv.x;
    o.y = tv.y * sn + bv.y;
    o.z = tv.z * sn + bv.z;
    o.w = tv.w * sn + bv.w;
    agg4[i] = o;
}

// ---------------------------------------------------------------------------
// Edge scatter: one wave per edge, lane handles 4 features.
// agg[dst] += t[src] * (dis[src]*dis[dst])   via global_atomic_add_f32 (L2).
// t and agg are L2-resident (25.6 MB each vs 192 MB L2) -> no HBM round trips.
// ---------------------------------------------------------------------------
__global__ void k_scatter(float* __restrict__ agg, const float* __restrict__ t,
                          const float* __restrict__ dis,
                          const int* __restrict__ src, const int* __restrict__ dst,
                          int n_edges) {
    int e    = blockIdx.x * (blockDim.x >> 5) + (threadIdx.x >> 5);
    int lane = threadIdx.x & 31;
    if (e >= n_edges) return;
    int s = src[e];
    int d = dst[e];
    float en = dis[s] * dis[d];
    float4 v = ((const float4*)(t + (size_t)s * HID))[lane];
    float* a = agg + (size_t)d * HID + lane * 4;
    atomicAdd(a + 0, v.x * en);
    atomicAdd(a + 1, v.y * en);
    atomicAdd(a + 2, v.z * en);
    atomicAdd(a + 3, v.w * en);
}

// ---------------------------------------------------------------------------
// Per-feature sum / sum-of-squares partials (BatchNorm stats).
// 256 threads: feature f = tid&127, row stride 2. Atomic merge into sums[128].
// ---------------------------------------------------------------------------
__global__ void k_stats(const float* __restrict__ agg, float* __restrict__ sums,
                        float* __restrict__ sumsq, int rows_per_block, int n_rows) {
    int f    = threadIdx.x & 127;
    int rsub = threadIdx.x >> 7;
    int base = blockIdx.x * rows_per_block;
    int end  = base + rows_per_block;
    if (end > n_rows) end = n_rows;
    float s = 0.0f, q = 0.0f;
    for (int r = base + rsub; r < end; r += 2) {
        float v = agg[(size_t)r * HID + f];
        s += v;
        q += v * v;
    }
    atomicAdd(&sums[f], s);
    atomicAdd(&sumsq[f], q);
}

// ---------------------------------------------------------------------------
// Fused BN (biased var) + optional skip + ReLU; h = result; acc (+)= h.
// ---------------------------------------------------------------------------
__global__ void k_norm(float* __restrict__ h, float* __restrict__ acc,
                       const float* __restrict__ agg, const float* __restrict__ sums,
                       const float* __restrict__ sumsq, const float* __restrict__ gamma,
                       const float* __restrict__ beta, const float* __restrict__ xskip,
                       int do_skip, int init_acc, int total) {
    int i = blockIdx.x * blockDim.x + threadIdx.x;
    if (i >= total) return;
    int f = i & 127;
    const float inv_n = 1.0f / (float)N_NODES;
    float mean = sums[f] * inv_n;
    float var  = sumsq[f] * inv_n - mean * mean;
    float t = (agg[i] - mean) * __frsqrt_rn(var + 1e-5f) * gamma[f] + beta[f];
    if (do_skip) t += 0.5f * xskip[i];
    t = fmaxf(t, 0.0f);
    h[i] = t;
    acc[i] = init_acc ? t : (acc[i] + t);
}

// ---------------------------------------------------------------------------
// Launch
// ---------------------------------------------------------------------------
extern "C" void kernel_launch(void* const* d_in, const int* in_sizes, int n_in,
                              void* d_out, int out_size, void* d_ws, size_t ws_size,
                              hipStream_t stream) {
    (void)in_sizes; (void)n_in; (void)out_size; (void)ws_size;

    const float* x     = (const float*)d_in[0];
    const float* W_in  = (const float*)d_in[1];
    const float* b_in  = (const float*)d_in[2];
    const float* W_g   = (const float*)d_in[3];
    const float* b_g   = (const float*)d_in[4];
    const float* gamma = (const float*)d_in[5];
    const float* beta  = (const float*)d_in[6];
    const float* W_out = (const float*)d_in[7];
    const float* b_out = (const float*)d_in[8];
    const int*   edge  = (const int*)d_in[9];
    const int* src = edge;
    const int* dst = edge + N_EDGES;
    float* out = (float*)d_out;

    float* ws    = (float*)d_ws;
    float* xskip = ws;                 // [NF]
    float* h     = xskip + NF;         // [NF]
    float* t     = h + NF;             // [NF]
    float* agg   = t + NF;             // [NF]
    float* acc   = agg + NF;           // [NF]
    float* dis   = acc + NF;           // [N_NODES] deg -> rsqrt in place
    float* sums  = dis + N_NODES;      // [128]
    float* sumsq = sums + HID;         // [128]

    const int T = 256;

    // degree + normalization
    k_fill<<<(N_NODES + T - 1) / T, T, 0, stream>>>(dis, N_NODES, 0.0f);
    k_deg<<<(N_EDGES + T - 1) / T, T, 0, stream>>>(dis, dst, N_EDGES);
    k_dis<<<(N_NODES + T - 1) / T, T, 0, stream>>>(dis, N_NODES);

    // h0 = x @ W_in + b_in  -> xskip
    k_gemm<HID><<<N_NODES / 16, 256, 0, stream>>>(xskip, x, W_in, b_in);

    const int total4 = NF / 4;
    for (int i = 0; i < N_GCN; ++i) {
        const float* hin = (i == 0) ? xskip : h;
        // t = h @ W_g[i]
        k_gemm<HID><<<N_NODES / 16, 256, 0, stream>>>(t, hin, W_g + (size_t)i * HID * HID, nullptr);
        // agg = t * self_norm + b_g[i]
        k_agg_init<<<(total4 + T - 1) / T, T, 0, stream>>>((float4*)agg, (const float4*)t,
                                                           dis, b_g + (size_t)i * HID, total4);
        // agg[dst] += t[src] * enorm
        k_scatter<<<(N_EDGES * 32 + T - 1) / T, T, 0, stream>>>(agg, t, dis, src, dst, N_EDGES);
        // batch-norm stats
        k_fill<<<1, 256, 0, stream>>>(sums, 2 * HID, 0.0f);
        k_stats<<<200, 256, 0, stream>>>(agg, sums, sumsq, 250, N_NODES);
        // normalize + skip + relu + accumulate
        k_norm<<<(NF + T - 1) / T, T, 0, stream>>>(h, acc, agg, sums, sumsq,
                                                   gamma + (size_t)i * HID, beta + (size_t)i * HID,
                                                   xskip, ((i + 1) % 2 == 0) ? 1 : 0,
                                                   (i == 0) ? 1 : 0, NF);
    }

    // out = acc @ W_out + b_out
    k_gemm<OUT_C><<<N_NODES / 16, 128, 0, stream>>>(out, acc, W_out, b_out);
}